// EchoStateNetwork_14989435863543
// MI455X (gfx1250) — compile-verified
//
#include <hip/hip_runtime.h>
#include <math.h>

// Echo-state network scan on gfx1250 (MI455X).
// h_t = 0.1*h_{t-1} + 0.9*tanh(x_t@W_in + h_{t-1}@W_rec + b)
// B=64, T=512, I=32, R=2048.  One kernel launch per timestep (grid-wide
// dependency between steps); recurrent GEMM uses V_WMMA_F32_16X16X4_F32
// (full fp32 matrix path).  LDS staging is double-buffered so the global
// loads + ds_stores of chunk c+1 co-execute with the WMMAs of chunk c:
// one barrier per K-chunk on the serial critical path.

#define ESN_B 64
#define ESN_T 512
#define ESN_I 32
#define ESN_R 2048
#define NBLK  64     // N columns per workgroup
#define KCH   32     // K chunk staged through LDS per iteration
#define NCHUNK (ESN_R / KCH)
#define SAP   34     // padded LDS row stride (dwords), A-side tiles ([m][k] row-major)
#define SBP   34     // padded LDS row stride (dwords), B-side tiles (transposed [n][k])

typedef float v2f __attribute__((ext_vector_type(2)));
typedef float v8f __attribute__((ext_vector_type(8)));

__device__ __forceinline__ v8f wmma4(v2f a, v2f b, v8f c) {
  // D = A(16x4,f32) * B(4x16,f32) + C(16x16,f32)
  return __builtin_amdgcn_wmma_f32_16x16x4_f32(
      /*neg_a=*/false, a, /*neg_b=*/false, b,
      /*c_mod=*/(short)0, c, /*reuse_a=*/false, /*reuse_b=*/false);
}

__device__ __forceinline__ float esn_tanh(float v) {
#if __has_builtin(__builtin_amdgcn_tanhf)
  return __builtin_amdgcn_tanhf(v);          // gfx1250 V_TANH_F32 trans op
#elif __has_builtin(__builtin_amdgcn_tanh_f32)
  return __builtin_amdgcn_tanh_f32(v);
#else
  return tanhf(v);                           // ocml fallback
#endif
}

// Stage one K-chunk: h_{t-1}[0:64, k0:k0+32] row-major into sAbuf, and
// W_rec[k0:k0+32, n0:n0+64] transposed ([n][k]) into sBbuf.
__device__ __forceinline__ void stage_chunk(const float* __restrict__ hprev,
                                            const float* __restrict__ Wrec,
                                            float* sAbuf, float* sBbuf,
                                            int tid, int k0, int n0, size_t TR) {
  {
    const int b  = tid >> 2;                  // 0..63
    const int kk = (tid & 3) * 8;             // 0,8,16,24
    const float4* src = reinterpret_cast<const float4*>(
        hprev + (size_t)b * TR + k0 + kk);
    float4 v0 = src[0], v1 = src[1];
    float* dst = &sAbuf[b * SAP + kk];
    dst[0]=v0.x; dst[1]=v0.y; dst[2]=v0.z; dst[3]=v0.w;
    dst[4]=v1.x; dst[5]=v1.y; dst[6]=v1.z; dst[7]=v1.w;
  }
  {
    const int k  = tid >> 3;                  // 0..31
    const int nn = (tid & 7) * 8;             // 0..56
    const float4* src = reinterpret_cast<const float4*>(
        Wrec + (size_t)(k0 + k) * ESN_R + n0 + nn);
    float4 v0 = src[0], v1 = src[1];
    const float tmp[8] = {v0.x,v0.y,v0.z,v0.w,v1.x,v1.y,v1.z,v1.w};
#pragma unroll
    for (int j = 0; j < 8; ++j) sBbuf[(nn + j) * SBP + k] = tmp[j];
  }
}

__global__ __launch_bounds__(256)
void esn_step_kernel(const float* __restrict__ x,     // [B,T,I]
                     const float* __restrict__ Win,   // [I,R]
                     const float* __restrict__ Wrec,  // [R,R]
                     const float* __restrict__ bias,  // [R]
                     float* __restrict__ out,         // [B,T,R]
                     int t)
{
  __shared__ float sA [2][ESN_B * SAP];  // h_{t-1} chunk, double-buffered
  __shared__ float sB [2][NBLK  * SBP];  // W_rec chunk (transposed), double-buffered
  __shared__ float sX [ESN_B * SAP];     // x_t, [64][32] row-major
  __shared__ float sWt[NBLK  * SBP];     // W_in block, transposed [n][k]

  const int tid  = threadIdx.x;
  const int lane = tid & 31;
  const int wave = tid >> 5;         // 0..7
  const int mt   = wave & 3;         // M tile: rows mt*16 .. mt*16+15 of B=64
  const int ng   = wave >> 2;        // 0..1 -> owns N tiles {2ng, 2ng+1}
  const int lr   = lane & 15;
  const int hi   = lane >> 4;        // 0/1: second half-wave holds K+2/K+3
  const int n0   = blockIdx.x * NBLK;
  const size_t TR = (size_t)ESN_T * ESN_R;

  // ---- stage x_t and the W_in block once per step ----
  {
    const int b  = tid >> 2;
    const int i0 = (tid & 3) * 8;
    const float4* src = reinterpret_cast<const float4*>(
        x + (size_t)b * ESN_T * ESN_I + (size_t)t * ESN_I + i0);
    float4 v0 = src[0], v1 = src[1];
    float* dst = &sX[b * SAP + i0];
    dst[0]=v0.x; dst[1]=v0.y; dst[2]=v0.z; dst[3]=v0.w;
    dst[4]=v1.x; dst[5]=v1.y; dst[6]=v1.z; dst[7]=v1.w;
  }
  {
    const int k  = tid >> 3;
    const int nn = (tid & 7) * 8;
    const float4* src = reinterpret_cast<const float4*>(
        Win + (size_t)k * ESN_R + n0 + nn);
    float4 v0 = src[0], v1 = src[1];
    const float tmp[8] = {v0.x,v0.y,v0.z,v0.w,v1.x,v1.y,v1.z,v1.w};
#pragma unroll
    for (int j = 0; j < 8; ++j) sWt[(nn + j) * SBP + k] = tmp[j];
  }

  v8f acc0 = {};
  v8f acc1 = {};

  const int an = mt * 16 + lr;                 // A-fragment row
  const int bn0 = (ng * 2 + 0) * 16 + lr;      // B-fragment cols (transposed rows)
  const int bn1 = (ng * 2 + 1) * 16 + lr;

  // ---- recurrent GEMM: h_{t-1} @ W_rec, K = 2048 in chunks of 32 ----
  if (t > 0) {
    const float* hprev = out + (size_t)(t - 1) * ESN_R;   // row b at +b*TR

    stage_chunk(hprev, Wrec, sA[0], sB[0], tid, 0, n0, TR);

    for (int c = 0; c < NCHUNK; ++c) {
      __syncthreads();                // buf cur ready; buf cur^1 free (WAR safe)
      const int cur = c & 1;
      if (c + 1 < NCHUNK)
        stage_chunk(hprev, Wrec, sA[cur ^ 1], sB[cur ^ 1], tid,
                    (c + 1) * KCH, n0, TR);

      const float* cA = sA[cur];
      const float* cB = sB[cur];
#pragma unroll
      for (int kk = 0; kk < KCH; kk += 4) {
        const int kf = kk + 2 * hi;   // lanes 0-15: K={k,k+1}; 16-31: K={k+2,k+3}
        v2f a  = *(const v2f*)&cA[an  * SAP + kf];
        v2f b0 = *(const v2f*)&cB[bn0 * SBP + kf];
        v2f b1 = *(const v2f*)&cB[bn1 * SBP + kf];
        acc0 = wmma4(a, b0, acc0);
        acc1 = wmma4(a, b1, acc1);
      }
    }
  }

  // ---- input projection: x_t @ W_in, K = 32 ----
  __syncthreads();
#pragma unroll
  for (int kk = 0; kk < ESN_I; kk += 4) {
    const int kf = kk + 2 * hi;
    v2f a  = *(const v2f*)&sX [an  * SAP + kf];
    v2f b0 = *(const v2f*)&sWt[bn0 * SBP + kf];
    v2f b1 = *(const v2f*)&sWt[bn1 * SBP + kf];
    acc0 = wmma4(a, b0, acc0);
    acc1 = wmma4(a, b1, acc1);
  }

  // ---- epilogue: bias + tanh + leaky blend, write column t ----
  const float aLk = 0.9f, omA = 0.1f;
#pragma unroll
  for (int r = 0; r < 8; ++r) {
    const int Mg = mt * 16 + r + 8 * hi;               // C/D layout: M = r + 8*hi
    float* orow  = out + (size_t)Mg * TR + (size_t)t * ESN_R;
    const int Ng0 = n0 + (ng * 2 + 0) * 16 + lr;
    const int Ng1 = n0 + (ng * 2 + 1) * 16 + lr;
    float hp0 = 0.f, hp1 = 0.f;
    if (t > 0) {
      const float* prow = orow - ESN_R;                // column t-1
      hp0 = prow[Ng0];
      hp1 = prow[Ng1];
    }
    orow[Ng0] = omA * hp0 + aLk * esn_tanh(acc0[r] + bias[Ng0]);
    orow[Ng1] = omA * hp1 + aLk * esn_tanh(acc1[r] + bias[Ng1]);
  }
}

extern "C" void kernel_launch(void* const* d_in, const int* in_sizes, int n_in,
                              void* d_out, int out_size, void* d_ws, size_t ws_size,
                              hipStream_t stream) {
  (void)in_sizes; (void)n_in; (void)out_size; (void)d_ws; (void)ws_size;
  const float* x    = (const float*)d_in[0];
  const float* Win  = (const float*)d_in[1];
  const float* Wrec = (const float*)d_in[2];
  const float* bias = (const float*)d_in[3];
  float* out        = (float*)d_out;

  dim3 grid(ESN_R / NBLK);   // 32 workgroups per step
  dim3 block(256);           // 8 wave32s
  for (int t = 0; t < ESN_T; ++t) {
    esn_step_kernel<<<grid, block, 0, stream>>>(x, Win, Wrec, bias, out, t);
  }
}